// GCN_10033043603648
// MI455X (gfx1250) — compile-verified
//
#include <hip/hip_runtime.h>

typedef __attribute__((ext_vector_type(2))) float v2f;
typedef __attribute__((ext_vector_type(8))) float v8f;

#define TPB 256

// ---------------------------------------------------------------- init
__global__ void gcn_init(float* __restrict__ deg, int* __restrict__ cursor,
                         int* __restrict__ counter, float* __restrict__ gsum,
                         float* __restrict__ gcnt, int N, int G) {
  int i = blockIdx.x * blockDim.x + threadIdx.x;
  if (i < N) {
    deg[i] = 1.0f;   // self-loop contribution
    cursor[i] = 0;
  }
  if (i < G * 32) gsum[i] = 0.0f;
  if (i < G) gcnt[i] = 0.0f;
  if (i == 0) *counter = 0;
}

// ---------------------------------------------------------------- degree count (float, also feeds dinv)
__global__ void gcn_deg(const int* __restrict__ col, float* __restrict__ deg, int E) {
  int e = blockIdx.x * blockDim.x + threadIdx.x;
  if (e < E) atomicAdd(&deg[col[e]], 1.0f);
}

// region allocation: CSR regions need not be in node order, so a global bump
// allocator replaces a prefix scan. cnt recovered exactly from float degree.
__global__ void gcn_alloc(const float* __restrict__ deg, int* __restrict__ cnt,
                          int* __restrict__ start, int* __restrict__ counter, int N) {
  int i = blockIdx.x * blockDim.x + threadIdx.x;
  if (i < N) {
    int n = (int)(deg[i] + 0.5f) - 1;  // in-degree without self loop
    cnt[i] = n;
    start[i] = atomicAdd(counter, n);
  }
}

__global__ void gcn_dinv(float* __restrict__ d, int N) {
  int i = blockIdx.x * blockDim.x + threadIdx.x;
  if (i < N) d[i] = rsqrtf(d[i]);  // deg >= 1 always (self-loops)
}

// slot placement: store source node id per incoming edge of col[e]
__global__ void gcn_fill(const int* __restrict__ row, const int* __restrict__ col,
                         const int* __restrict__ start, int* __restrict__ cursor,
                         int* __restrict__ elist, int E) {
  int e = blockIdx.x * blockDim.x + threadIdx.x;
  if (e < E) {
    int c = col[e];
    int p = atomicAdd(&cursor[c], 1);
    elist[start[c] + p] = row[e];
  }
}

// ---------------------------------------------------------------- fp32 WMMA GEMM
// Out[N x H] = act(In)[N x K] @ W[K x H], act = optional (+bias, relu) on input.
template <int K, bool RELU_BIAS>
__global__ void gcn_gemm_wmma(const float* __restrict__ In, const float* __restrict__ W,
                              const float* __restrict__ bias, float* __restrict__ Out,
                              int Nrows, int H, int rowTilesPerBlock) {
  const int lane = threadIdx.x & 31;
  const int half = lane >> 4;   // 0: lanes 0-15, 1: lanes 16-31
  const int l16  = lane & 15;
  const int colBase = blockIdx.y * 16;
  constexpr int KQ = K / 4;     // number of 16x16x4 k-steps

  // B tile (K x 16): v0 holds K=k0+2*half, v1 holds K=k0+2*half+1, N=l16
  v2f b[KQ];
#pragma unroll
  for (int s = 0; s < KQ; ++s) {
    const int k = s * 4 + 2 * half;
    b[s].x = W[(k + 0) * H + colBase + l16];
    b[s].y = W[(k + 1) * H + colBase + l16];
  }

  for (int rt = 0; rt < rowTilesPerBlock; ++rt) {
    const int rowBase = (blockIdx.x * rowTilesPerBlock + rt) * 16;
    if (rowBase >= Nrows) return;
    const float* arow = In + (size_t)(rowBase + l16) * K;
    v8f c = {};
#pragma unroll
    for (int s = 0; s < KQ; ++s) {
      const int k = s * 4 + 2 * half;
      v2f a;  // A 16x4 layout: lane l16 = row M, v0/v1 = K=k0+2*half(+1)
      if (RELU_BIAS) {
        float a0 = arow[k + 0] + bias[k + 0];
        float a1 = arow[k + 1] + bias[k + 1];
        a.x = a0 > 0.0f ? a0 : 0.0f;
        a.y = a1 > 0.0f ? a1 : 0.0f;
      } else {
        a = *(const v2f*)(arow + k);
      }
      c = __builtin_amdgcn_wmma_f32_16x16x4_f32(false, a, false, b[s], (short)0, c,
                                                false, false);
    }
    // C/D layout: VGPR v, lanes 0-15 -> M=v; lanes 16-31 -> M=v+8; N=l16
    float* orow = Out + (size_t)rowBase * H + colBase + l16;
#pragma unroll
    for (int v = 0; v < 8; ++v) orow[(size_t)(v + 8 * half) * H] = c[v];
  }
}

// ---------------------------------------------------------------- gather aggregation (no atomics)
// one wave per target node; lane covers a float4 of the 128-wide feature row
__global__ void gcn_gather128(const int* __restrict__ elist, const int* __restrict__ start,
                              const int* __restrict__ cnt, const float* __restrict__ dinv,
                              const float* __restrict__ h, float* __restrict__ agg, int N) {
  int gid = blockIdx.x * blockDim.x + threadIdx.x;
  int i = gid >> 5;
  int lane = gid & 31;
  if (i >= N) return;
  const float dc = dinv[i];
  const float4* hv = (const float4*)h;
  // self-loop term seeds the accumulator
  float4 s = hv[(size_t)i * 32 + lane];
  float sl = dc * dc;
  float ax = s.x * sl, ay = s.y * sl, az = s.z * sl, aw = s.w * sl;
  const int base = start[i];
  const int n = cnt[i];
  for (int j = 0; j < n; ++j) {
    int r = elist[base + j];
    float nrm = dinv[r] * dc;
    float4 v = hv[(size_t)r * 32 + lane];
    ax += v.x * nrm;
    ay += v.y * nrm;
    az += v.z * nrm;
    aw += v.w * nrm;
  }
  float4 o = make_float4(ax, ay, az, aw);
  ((float4*)agg)[(size_t)i * 32 + lane] = o;
}

// one wave per target node; lane = feature (F = 32)
__global__ void gcn_gather32(const int* __restrict__ elist, const int* __restrict__ start,
                             const int* __restrict__ cnt, const float* __restrict__ dinv,
                             const float* __restrict__ h, float* __restrict__ agg, int N) {
  int gid = blockIdx.x * blockDim.x + threadIdx.x;
  int i = gid >> 5;
  int lane = gid & 31;
  if (i >= N) return;
  const float dc = dinv[i];
  float acc = h[(size_t)i * 32 + lane] * dc * dc;  // self loop
  const int base = start[i];
  const int n = cnt[i];
  for (int j = 0; j < n; ++j) {
    int r = elist[base + j];
    acc += h[(size_t)r * 32 + lane] * (dinv[r] * dc);
  }
  agg[(size_t)i * 32 + lane] = acc;
}

// ---------------------------------------------------------------- pooling (fused relu(agg+b2))
__global__ void gcn_pool(const float* __restrict__ agg, const float* __restrict__ b2,
                         const int* __restrict__ batch, float* __restrict__ gsum,
                         float* __restrict__ gcnt, int N) {
  int gid = blockIdx.x * blockDim.x + threadIdx.x;
  int i = gid >> 5;
  int f = gid & 31;
  if (i >= N) return;
  int g = batch[i];
  float v = agg[(size_t)i * 32 + f] + b2[f];
  v = v > 0.0f ? v : 0.0f;
  atomicAdd(&gsum[g * 32 + f], v);
  if (f == 0) atomicAdd(&gcnt[g], 1.0f);
}

// ---------------------------------------------------------------- MLP head per graph
__global__ void gcn_head(const float* __restrict__ gsum, const float* __restrict__ gcnt,
                         const float* __restrict__ fw1, const float* __restrict__ fb1,
                         const float* __restrict__ fw2, const float* __restrict__ fb2,
                         float* __restrict__ out) {
  __shared__ float sm[32];
  __shared__ float sz[16];
  int g = blockIdx.x;
  int t = threadIdx.x;
  float c = gcnt[g];
  c = c > 1.0f ? c : 1.0f;
  sm[t] = gsum[g * 32 + t] / c;
  __syncthreads();
  if (t < 16) {
    float z = fb1[t];
#pragma unroll
    for (int f = 0; f < 32; ++f) z += sm[f] * fw1[f * 16 + t];
    sz[t] = tanhf(z);
  }
  __syncthreads();
  if (t == 0) {
    float o = fb2[0];
#pragma unroll
    for (int j = 0; j < 16; ++j) o += sz[j] * fw2[j];
    out[g] = 1.0f / (1.0f + expf(-o));
  }
}

// ---------------------------------------------------------------- launch
extern "C" void kernel_launch(void* const* d_in, const int* in_sizes, int n_in,
                              void* d_out, int out_size, void* d_ws, size_t ws_size,
                              hipStream_t stream) {
  const float* x   = (const float*)d_in[0];
  const float* W1  = (const float*)d_in[1];
  const float* b1  = (const float*)d_in[2];
  const float* W2  = (const float*)d_in[3];
  const float* b2  = (const float*)d_in[4];
  const float* fw1 = (const float*)d_in[5];
  const float* fb1 = (const float*)d_in[6];
  const float* fw2 = (const float*)d_in[7];
  const float* fb2 = (const float*)d_in[8];
  const int* eidx  = (const int*)d_in[9];
  const int* batch = (const int*)d_in[10];

  const int N = in_sizes[10];       // batch has one id per node
  const int E = in_sizes[9] / 2;    // edge_index is [2, E]
  const int G = out_size;           // output is [G, 1]
  const int H1 = 128, H2 = 32;

  const int* row = eidx;            // sources (gather side)
  const int* col = eidx + E;        // targets (aggregation side)

  // workspace carving (1KB-aligned chunks)
  char* ws = (char*)d_ws;
  size_t off = 0;
  auto carve = [&](size_t bytes) {
    void* p = ws + off;
    off += (bytes + 1023) & ~(size_t)1023;
    return p;
  };
  float* dinv   = (float*)carve((size_t)N * 4);        // deg then dinv (in place)
  int*   cnt    = (int*)carve((size_t)N * 4);
  int*   start  = (int*)carve((size_t)N * 4);
  int*   cursor = (int*)carve((size_t)N * 4);
  int*   counter= (int*)carve(4);
  int*   elist  = (int*)carve((size_t)E * 4);
  float* bufA   = (float*)carve((size_t)N * 128 * 4);  // h1, later t
  float* bufB   = (float*)carve((size_t)N * 128 * 4);  // agg1, later agg2
  float* gsum   = (float*)carve((size_t)G * 32 * 4);
  float* gcnt   = (float*)carve((size_t)G * 4);
  (void)ws_size; (void)n_in;

  // degrees / CSR build / normalization
  {
    int total = (N > G * 32) ? N : G * 32;
    gcn_init<<<(total + TPB - 1) / TPB, TPB, 0, stream>>>(dinv, cursor, counter,
                                                          gsum, gcnt, N, G);
  }
  gcn_deg<<<(E + TPB - 1) / TPB, TPB, 0, stream>>>(col, dinv, E);
  gcn_alloc<<<(N + TPB - 1) / TPB, TPB, 0, stream>>>(dinv, cnt, start, counter, N);
  gcn_dinv<<<(N + TPB - 1) / TPB, TPB, 0, stream>>>(dinv, N);
  gcn_fill<<<(E + TPB - 1) / TPB, TPB, 0, stream>>>(row, col, start, cursor, elist, E);

  const int RT = 8;                                  // row tiles per block
  const int rowTiles = (N + 15) / 16;
  const unsigned gbx = (unsigned)((rowTiles + RT - 1) / RT);
  const unsigned nodeWaves = (unsigned)(((size_t)N * 32 + TPB - 1) / TPB);

  // layer 1: h1 = x @ W1, then agg1 = A_norm @ h1 (gather)
  gcn_gemm_wmma<128, false><<<dim3(gbx, H1 / 16), 32, 0, stream>>>(
      x, W1, nullptr, bufA, N, H1, RT);
  gcn_gather128<<<nodeWaves, TPB, 0, stream>>>(elist, start, cnt, dinv, bufA, bufB, N);

  // layer 2: t = relu(agg1 + b1) @ W2 (bias+relu fused into A load), then gather
  gcn_gemm_wmma<128, true><<<dim3(gbx, H2 / 16), 32, 0, stream>>>(
      bufB, W2, b1, bufA, N, H2, RT);
  gcn_gather32<<<nodeWaves, TPB, 0, stream>>>(elist, start, cnt, dinv, bufA, bufB, N);

  // pool (fused relu(agg2 + b2)) and head
  gcn_pool<<<nodeWaves, TPB, 0, stream>>>(bufB, b2, batch, gsum, gcnt, N);
  gcn_head<<<G, 32, 0, stream>>>(gsum, gcnt, fw1, fb1, fw2, fb2, (float*)d_out);
}